// DynamicRelationConv_59820304498979
// MI455X (gfx1250) — compile-verified
//
#include <hip/hip_runtime.h>
#include <cstdint>

#define C 128
#define FLT_BIG 3.402823466e38f

typedef float v2f __attribute__((ext_vector_type(2)));
typedef float v8f __attribute__((ext_vector_type(8)));

// ---------------- zero h ----------------
__global__ void k_zero(float* __restrict__ p, long long n) {
  long long i = (long long)blockIdx.x * blockDim.x + threadIdx.x;
  if (i < n) p[i] = 0.0f;
}

// ---------------- init aggregation buffers ----------------
__global__ void k_init_aggr(float* __restrict__ sum, float* __restrict__ mx,
                            float* __restrict__ mn, float* __restrict__ cnt,
                            int n_nodes) {
  long long i = (long long)blockIdx.x * blockDim.x + threadIdx.x;
  long long nc = (long long)n_nodes * C;
  if (i < nc) {
    sum[i] = 0.0f;
    mx[i]  = -FLT_BIG;
    mn[i]  =  FLT_BIG;
  }
  if (i < n_nodes) cnt[i] = 0.0f;
}

// ---------------- edge scatter: 128 lanes per edge (coalesced channels) ----
__global__ void k_scatter(const float* __restrict__ x, const int* __restrict__ ei,
                          int n_edges,
                          float* __restrict__ sum, float* __restrict__ mx,
                          float* __restrict__ mn, float* __restrict__ cnt) {
  long long tid = (long long)blockIdx.x * blockDim.x + threadIdx.x;
  int e = (int)(tid >> 7);
  int c = (int)(tid & 127);
  if (e >= n_edges) return;
  int s = ei[e];             // edge_index[0][e]  (source)
  int d = ei[n_edges + e];   // edge_index[1][e]  (destination)
  float v = x[(long long)s * C + c];
  long long o = (long long)d * C + c;
  __hip_atomic_fetch_add(&sum[o], v, __ATOMIC_RELAXED, __HIP_MEMORY_SCOPE_AGENT);
  __hip_atomic_fetch_max(&mx[o],  v, __ATOMIC_RELAXED, __HIP_MEMORY_SCOPE_AGENT);
  __hip_atomic_fetch_min(&mn[o],  v, __ATOMIC_RELAXED, __HIP_MEMORY_SCOPE_AGENT);
  if (c == 0)
    __hip_atomic_fetch_add(&cnt[d], 1.0f, __ATOMIC_RELAXED, __HIP_MEMORY_SCOPE_AGENT);
}

// ---------------- finalize: mean, zero empty-segment max/min ----------------
__global__ void k_finalize(const float* __restrict__ sum, float* __restrict__ mx,
                           float* __restrict__ mn, float* __restrict__ mean,
                           const float* __restrict__ cnt, int n_nodes) {
  long long i = (long long)blockIdx.x * blockDim.x + threadIdx.x;
  long long nc = (long long)n_nodes * C;
  if (i >= nc) return;
  int node = (int)(i >> 7);
  float cn = cnt[node];
  mean[i] = sum[i] / fmaxf(cn, 1.0f);
  if (cn == 0.0f) { mx[i] = 0.0f; mn[i] = 0.0f; }
}

// ---------------- WMMA GEMM ----------------
// Each wave computes a full 16-row x 128-col strip of h: 8 v8f accumulators.
// Per K-step: 1 A-fragment load (reused 8x) + 8 B-fragment loads + 8 WMMAs.
// A (M=16,K=4) f32: lanes 0-15 hold K={0,1}, lanes 16-31 hold K={2,3}.
// C/D 16x16 f32: VGPR v -> row v (lanes 0-15) / row v+8 (lanes 16-31).
__global__ __launch_bounds__(128) void k_gemm_wmma(
    const float* __restrict__ x,
    const float* __restrict__ mean, const float* __restrict__ mx,
    const float* __restrict__ mn,   const float* __restrict__ sum,
    const float* __restrict__ W1,   const float* __restrict__ W2,
    float* __restrict__ h, int n_nodes) {
  const int lane = threadIdx.x & 31;
  const int wv   = threadIdx.x >> 5;              // 4 waves per block
  const int m0   = blockIdx.x * 64 + wv * 16;     // each wave: one 16-row strip
  if (m0 >= n_nodes) return;                      // wave-uniform exit

  int row = m0 + (lane & 15);
  if (row >= n_nodes) row = n_nodes - 1;          // tail clamp, EXEC stays all-1s
  const int khalf = (lane >> 4) << 1;             // 0 or 2
  const int nlane = lane & 15;

  v8f acc[8] = {};

  const float* Asrc[5] = { x, mean, mx, mn, sum };
#pragma unroll
  for (int seg = 0; seg < 5; ++seg) {
    const float* A   = Asrc[seg];
    const float* W   = (seg == 0) ? W1 : W2;
    const int    ldw = (seg == 0) ? C : 4 * C;
    const int    wk  = (seg == 0) ? 0 : (seg - 1) * C;
    const float* Ap  = A + (long long)row * C;
    const float* Wp  = W + (long long)nlane * ldw + wk;
#pragma unroll 2
    for (int k0 = 0; k0 < C; k0 += 4) {
      const int kk = k0 + khalf;
      v2f a = *(const v2f*)(Ap + kk);
#pragma unroll
      for (int j = 0; j < 8; ++j) {               // 8 n-tiles of 16 cols
        v2f b = *(const v2f*)(Wp + j * 16 * ldw + kk);
        acc[j] = __builtin_amdgcn_wmma_f32_16x16x4_f32(
            false, a, false, b, (short)0, acc[j], false, false);
      }
    }
  }

  // accumulate D tiles into h (wave exclusively owns its 16x128 strip)
  const int rbase = m0 + ((lane >> 4) << 3);
#pragma unroll
  for (int v = 0; v < 8; ++v) {
    int r = rbase + v;
    if (r < n_nodes) {
      float* hp = h + (long long)r * C + nlane;
#pragma unroll
      for (int j = 0; j < 8; ++j) hp[j * 16] += acc[j][v];
    }
  }
}

// ---------------- relu(0.5*h) + layernorm epilogue (wave32 reductions) ------
__global__ __launch_bounds__(128) void k_ln(const float* __restrict__ h,
                                            const float* __restrict__ gamma,
                                            const float* __restrict__ beta,
                                            float* __restrict__ out, int n_nodes) {
  int node = blockIdx.x;
  int c = threadIdx.x;
  float v = h[(long long)node * C + c];
  v = fmaxf(0.5f * v, 0.0f);

  float s = v, s2 = v * v;
#pragma unroll
  for (int off = 16; off > 0; off >>= 1) {   // wave32 butterfly
    s  += __shfl_xor(s,  off, 32);
    s2 += __shfl_xor(s2, off, 32);
  }
  __shared__ float ps[4], ps2[4];
  int wid = threadIdx.x >> 5;
  if ((threadIdx.x & 31) == 0) { ps[wid] = s; ps2[wid] = s2; }
  __syncthreads();
  float ts  = ps[0]  + ps[1]  + ps[2]  + ps[3];
  float ts2 = ps2[0] + ps2[1] + ps2[2] + ps2[3];
  float mu  = ts  * (1.0f / C);
  float var = ts2 * (1.0f / C) - mu * mu;
  out[(long long)node * C + c] = (v - mu) * rsqrtf(var + 1e-5f) * gamma[c] + beta[c];
}

// ---------------- launch -----------------------------------------------------
extern "C" void kernel_launch(void* const* d_in, const int* in_sizes, int n_in,
                              void* d_out, int out_size, void* d_ws, size_t ws_size,
                              hipStream_t stream) {
  const float* x     = (const float*)d_in[0];
  const int*   ei1   = (const int*)d_in[1];
  const int*   ei2   = (const int*)d_in[2];
  const float* W1_r1 = (const float*)d_in[3];
  const float* W2_r1 = (const float*)d_in[4];
  const float* W1_r2 = (const float*)d_in[5];
  const float* W2_r2 = (const float*)d_in[6];
  const float* gamma = (const float*)d_in[7];
  const float* beta  = (const float*)d_in[8];
  float* out = (float*)d_out;

  const int N = in_sizes[0] / C;
  const int E = in_sizes[1] / 2;
  const long long NC = (long long)N * C;

  float* sum  = (float*)d_ws;
  float* mx   = sum  + NC;
  float* mn   = mx   + NC;
  float* mean = mn   + NC;
  float* cnt  = mean + NC;
  float* h    = cnt  + N;

  const int T = 256;
  dim3 gNC((unsigned)((NC + T - 1) / T));
  dim3 gE((unsigned)(((long long)E * C + T - 1) / T));
  dim3 gM((unsigned)((N + 63) / 64));

  k_zero<<<gNC, T, 0, stream>>>(h, NC);

  // relation 1
  k_init_aggr<<<gNC, T, 0, stream>>>(sum, mx, mn, cnt, N);
  k_scatter<<<gE, T, 0, stream>>>(x, ei1, E, sum, mx, mn, cnt);
  k_finalize<<<gNC, T, 0, stream>>>(sum, mx, mn, mean, cnt, N);
  k_gemm_wmma<<<gM, 128, 0, stream>>>(x, mean, mx, mn, sum, W1_r1, W2_r1, h, N);

  // relation 2
  k_init_aggr<<<gNC, T, 0, stream>>>(sum, mx, mn, cnt, N);
  k_scatter<<<gE, T, 0, stream>>>(x, ei2, E, sum, mx, mn, cnt);
  k_finalize<<<gNC, T, 0, stream>>>(sum, mx, mn, mean, cnt, N);
  k_gemm_wmma<<<gM, 128, 0, stream>>>(x, mean, mx, mn, sum, W1_r2, W2_r2, h, N);

  k_ln<<<(unsigned)N, C, 0, stream>>>(h, gamma, beta, out, N);
}